// Euler_88304527606345
// MI455X (gfx1250) — compile-verified
//
#include <hip/hip_runtime.h>
#include <math.h>

typedef __attribute__((ext_vector_type(2))) float v2f;
typedef __attribute__((ext_vector_type(8))) float v8f;

#define NN   20000
#define EE   320000
#define TT   4
#define HID  128
#define G3   384   // 3*EMB

// ---------------------------------------------------------------------------
// elementwise helpers
// ---------------------------------------------------------------------------
__global__ void fill_zero_k(float* __restrict__ p, int n) {
    int i = blockIdx.x * blockDim.x + threadIdx.x;
    if (i < n) p[i] = 0.0f;
}

// deg counts over dst (per snapshot); eis layout [T,2,E]
__global__ void count_deg_k(const int* __restrict__ eis, float* __restrict__ deg) {
    int i = blockIdx.x * blockDim.x + threadIdx.x;
    if (i >= TT * EE) return;
    int t = i / EE, e = i - t * EE;
    int d = eis[(size_t)t * 2 * EE + EE + e];
    atomicAdd(&deg[t * NN + d], 1.0f);
}

__global__ void finalize_dinv_k(float* __restrict__ deg, int n) {
    int i = blockIdx.x * blockDim.x + threadIdx.x;
    if (i < n) deg[i] = rsqrtf(deg[i] + 1.0f);   // +1 = self loop; always > 0
}

// out[n][j] = bias[j] + H[n][j] * dinv[n]^2   (folds the self-loop edge)
__global__ void init_self_k(float* __restrict__ out, const float* __restrict__ H,
                            const float* __restrict__ dinv, const float* __restrict__ bias) {
    int i = blockIdx.x * blockDim.x + threadIdx.x;
    if (i >= NN * HID) return;
    int n = i >> 7, j = i & 127;
    float di = dinv[n];
    out[i] = bias[j] + H[i] * di * di;
}

// one edge handled by one wave32: lane covers 4 floats (float4 gather, 4 atomics)
__global__ void scatter_k(float* __restrict__ out, const float* __restrict__ H,
                          const float* __restrict__ dinv,
                          const int* __restrict__ src, const int* __restrict__ dst) {
    int i = blockIdx.x * blockDim.x + threadIdx.x;
    int e = i >> 5;
    if (e >= EE) return;
    int lane = i & 31;
    int s = src[e], d = dst[e];
    float norm = dinv[s] * dinv[d];
    const float4 v = *(const float4*)(H + (size_t)s * HID + lane * 4);
    float* o = out + (size_t)d * HID + lane * 4;
    atomicAdd(o + 0, v.x * norm);
    atomicAdd(o + 1, v.y * norm);
    atomicAdd(o + 2, v.z * norm);
    atomicAdd(o + 3, v.w * norm);
}

__global__ void relu_k(float* __restrict__ p, int n) {
    int i = blockIdx.x * blockDim.x + threadIdx.x;
    if (i < n) p[i] = fmaxf(p[i], 0.0f);
}

// ---------------------------------------------------------------------------
// fp32 WMMA GEMM:  C[M x Nout] = A[M x 128] * B  (+bias) (opt relu)
// B element (k, c) at B[k*rsB + c*csB].  M % 16 == 0, Nout % 64 == 0.
// Block = 256 thr = 8 waves; B panel (128 x 64) staged in LDS, pair-interleaved
// (stride 160 words -> b64 reads of both half-waves hit disjoint banks).
// Each wave owns TWO 16-row tiles sharing every B fetch.
// grid = (ceil(M/256), Nout/64)
// ---------------------------------------------------------------------------
#define BPS 160   // LDS words per K-pair row (64 cols * 2 + 32 pad)

__global__ void wmma_gemm_k(const float* __restrict__ A, const float* __restrict__ B,
                            const float* __restrict__ bias, float* __restrict__ C,
                            int Mrows, int Nout, int rsB, int csB, int relu) {
    __shared__ float Blds[64 * BPS];
    const int tid = threadIdx.x;
    const int col0 = blockIdx.y * 64;

    // stage B panel: element (k, c) -> Blds[(k/2)*BPS + c*2 + (k&1)]
    for (int idx = tid; idx < 128 * 64; idx += 256) {
        int k = idx >> 6, c = idx & 63;
        Blds[(k >> 1) * BPS + c * 2 + (k & 1)] =
            B[(size_t)k * rsB + (size_t)(col0 + c) * csB];
    }
    __syncthreads();

    const int wave = tid >> 5, lane = tid & 31;
    const int m = lane & 15;
    const int kof = (lane >> 4) * 2;             // lanes 16..31 hold K+2,K+3
    const int rt0 = blockIdx.x * 16 + wave;
    const int rt1 = rt0 + 8;
    const int r0 = rt0 * 16, r1 = rt1 * 16;
    // clamp OOB tiles to row 0 (results discarded at store)
    const float* A0 = A + (size_t)((r0 < Mrows ? r0 : 0) + m) * 128;
    const float* A1 = A + (size_t)((r1 < Mrows ? r1 : 0) + m) * 128;

    v8f acc0_0 = {}, acc0_1 = {}, acc0_2 = {}, acc0_3 = {};
    v8f acc1_0 = {}, acc1_1 = {}, acc1_2 = {}, acc1_3 = {};

    for (int k = 0; k < 128; k += 4) {
        int kk = k + kof;
        v2f a0 = *(const v2f*)(A0 + kk);
        v2f a1 = *(const v2f*)(A1 + kk);
        const float* bp = Blds + (kk >> 1) * BPS + m * 2;
        v2f b;
        b = *(const v2f*)(bp);
        acc0_0 = __builtin_amdgcn_wmma_f32_16x16x4_f32(false, a0, false, b, (short)0, acc0_0, false, false);
        acc1_0 = __builtin_amdgcn_wmma_f32_16x16x4_f32(false, a1, false, b, (short)0, acc1_0, false, false);
        b = *(const v2f*)(bp + 32);
        acc0_1 = __builtin_amdgcn_wmma_f32_16x16x4_f32(false, a0, false, b, (short)0, acc0_1, false, false);
        acc1_1 = __builtin_amdgcn_wmma_f32_16x16x4_f32(false, a1, false, b, (short)0, acc1_1, false, false);
        b = *(const v2f*)(bp + 64);
        acc0_2 = __builtin_amdgcn_wmma_f32_16x16x4_f32(false, a0, false, b, (short)0, acc0_2, false, false);
        acc1_2 = __builtin_amdgcn_wmma_f32_16x16x4_f32(false, a1, false, b, (short)0, acc1_2, false, false);
        b = *(const v2f*)(bp + 96);
        acc0_3 = __builtin_amdgcn_wmma_f32_16x16x4_f32(false, a0, false, b, (short)0, acc0_3, false, false);
        acc1_3 = __builtin_amdgcn_wmma_f32_16x16x4_f32(false, a1, false, b, (short)0, acc1_3, false, false);
    }

    // C/D layout: vgpr r, lanes 0-15 -> M=r, lanes 16-31 -> M=r+8 ; N = lane&15
    const int mb = (lane >> 4) * 8;
    float bv0 = bias ? bias[col0 + m]      : 0.0f;
    float bv1 = bias ? bias[col0 + 16 + m] : 0.0f;
    float bv2 = bias ? bias[col0 + 32 + m] : 0.0f;
    float bv3 = bias ? bias[col0 + 48 + m] : 0.0f;
    if (r0 < Mrows) {
#pragma unroll
        for (int r = 0; r < 8; ++r) {
            size_t ro = (size_t)(r0 + mb + r) * Nout;
            float v;
            v = acc0_0[r] + bv0; if (relu) v = fmaxf(v, 0.0f); C[ro + col0 + m]      = v;
            v = acc0_1[r] + bv1; if (relu) v = fmaxf(v, 0.0f); C[ro + col0 + 16 + m] = v;
            v = acc0_2[r] + bv2; if (relu) v = fmaxf(v, 0.0f); C[ro + col0 + 32 + m] = v;
            v = acc0_3[r] + bv3; if (relu) v = fmaxf(v, 0.0f); C[ro + col0 + 48 + m] = v;
        }
    }
    if (r1 < Mrows) {
#pragma unroll
        for (int r = 0; r < 8; ++r) {
            size_t ro = (size_t)(r1 + mb + r) * Nout;
            float v;
            v = acc1_0[r] + bv0; if (relu) v = fmaxf(v, 0.0f); C[ro + col0 + m]      = v;
            v = acc1_1[r] + bv1; if (relu) v = fmaxf(v, 0.0f); C[ro + col0 + 16 + m] = v;
            v = acc1_2[r] + bv2; if (relu) v = fmaxf(v, 0.0f); C[ro + col0 + 32 + m] = v;
            v = acc1_3[r] + bv3; if (relu) v = fmaxf(v, 0.0f); C[ro + col0 + 48 + m] = v;
        }
    }
}

// ---------------------------------------------------------------------------
// GRU scan: 20000 sequential steps, batch T=4 padded to M=16 for WMMA.
// One workgroup, 768 threads = 24 waves (one 16-col gh tile each, 384 cols).
// whh^T pair-interleaved in LDS (pair stride 800 -> conflict-free b64 reads);
// h padded to 16 rows of zeros (stride 132 -> conflict-free unconditional b64).
// ~219 KB LDS -- legal on CDNA5 (320KB/WGP).
// ---------------------------------------------------------------------------
#define WPS 800   // LDS words per K-pair row of whh^T (384 cols * 2 + 32 pad)
#define S_H 132

__device__ __forceinline__ float sigm(float x) { return 1.0f / (1.0f + __expf(-x)); }

__global__ void gru_scan_k(const float* __restrict__ GI, const float* __restrict__ whh,
                           const float* __restrict__ bhh, float* __restrict__ out) {
    extern __shared__ float smem[];
    float* whhT = smem;                         // 64 * WPS
    float* hS   = whhT + 64 * WPS;              // 16 * S_H (rows 4..15 stay zero)
    float* ghS  = hS + 16 * S_H;                // 4 * G3
    const int tid = threadIdx.x;

    // whh[j][k] -> whhT[(k/2)*WPS + j*2 + (k&1)]
    for (int idx = tid; idx < G3 * HID; idx += 768) {
        int j = idx >> 7, k = idx & 127;
        whhT[(k >> 1) * WPS + j * 2 + (k & 1)] = whh[idx];
    }
    for (int idx = tid; idx < 16 * S_H; idx += 768) hS[idx] = 0.0f;
    __syncthreads();

    const int wave = tid >> 5, lane = tid & 31;
    const int colb = wave * 16;                 // 24 waves * 16 = 384
    const int m    = lane & 15;
    const int kof  = (lane >> 4) * 2;

    for (int n = 0; n < NN; ++n) {
        // gh = h @ whh^T  (16x128x384, rows 0..3 live, 4..15 are zeros)
        v8f acc = {};
        for (int k = 0; k < 128; k += 4) {
            int kk = k + kof;
            v2f a = *(const v2f*)(hS + m * S_H + kk);
            v2f b = *(const v2f*)(whhT + (kk >> 1) * WPS + (colb + m) * 2);
            acc = __builtin_amdgcn_wmma_f32_16x16x4_f32(false, a, false, b, (short)0, acc, false, false);
        }
        if (lane < 16) {                        // rows 0..3 sit in vgprs 0..3, lanes 0..15
#pragma unroll
            for (int r = 0; r < 4; ++r) ghS[r * G3 + colb + lane] = acc[r];
        }
        __syncthreads();

        if (tid < 512) {                        // gates: (t, j) = (tid/128, tid%128)
            int t = tid >> 7, j = tid & 127;
            const float* gi = GI + ((size_t)t * NN + n) * G3;
            float hr = ghS[t * G3 + j]       + bhh[j];
            float hz = ghS[t * G3 + 128 + j] + bhh[128 + j];
            float hn = ghS[t * G3 + 256 + j] + bhh[256 + j];
            float r  = sigm(gi[j]       + hr);
            float z  = sigm(gi[128 + j] + hz);
            float nv = 2.0f * sigm(2.0f * (gi[256 + j] + r * hn)) - 1.0f;  // tanh
            float h  = hS[t * S_H + j];
            float h2 = (1.0f - z) * nv + z * h;
            hS[t * S_H + j] = h2;
            out[((size_t)t * NN + n) * HID + j] = h2;
        }
        __syncthreads();
    }
}

// ---------------------------------------------------------------------------
extern "C" void kernel_launch(void* const* d_in, const int* in_sizes, int n_in,
                              void* d_out, int out_size, void* d_ws, size_t ws_size,
                              hipStream_t stream) {
    const float* x   = (const float*)d_in[0];
    const int*   eis = (const int*)  d_in[1];
    const float* w1  = (const float*)d_in[2];
    const float* b1  = (const float*)d_in[3];
    const float* w2  = (const float*)d_in[4];
    const float* b2  = (const float*)d_in[5];
    const float* wih = (const float*)d_in[6];
    const float* whh = (const float*)d_in[7];
    const float* bih = (const float*)d_in[8];
    const float* bhh = (const float*)d_in[9];
    float* out = (float*)d_out;

    float* ws   = (float*)d_ws;
    float* XW1  = ws;                                   // N*HID
    float* dinv = XW1 + (size_t)NN * HID;               // T*N (deg then dinv)
    float* Z1   = dinv + (size_t)TT * NN;               // T*N*HID
    float* H2   = Z1 + (size_t)TT * NN * HID;           // T*N*HID
    float* GI   = H2 + (size_t)TT * NN * HID;           // T*N*G3
    float* Z2   = Z1;                                   // alias: Z1 dead after H2

    const int blk = 256;

    // degrees -> dinv
    fill_zero_k<<<(TT * NN + blk - 1) / blk, blk, 0, stream>>>(dinv, TT * NN);
    count_deg_k<<<(TT * EE + blk - 1) / blk, blk, 0, stream>>>(eis, dinv);
    finalize_dinv_k<<<(TT * NN + blk - 1) / blk, blk, 0, stream>>>(dinv, TT * NN);

    // XW1 = x @ w1   (t-invariant)
    wmma_gemm_k<<<dim3((NN / 16 + 15) / 16, HID / 64), blk, 0, stream>>>(
        x, w1, nullptr, XW1, NN, HID, HID, 1, 0);

    // layer 1: self-loop+bias init, edge scatter, relu
    for (int t = 0; t < TT; ++t) {
        init_self_k<<<(NN * HID + blk - 1) / blk, blk, 0, stream>>>(
            Z1 + (size_t)t * NN * HID, XW1, dinv + t * NN, b1);
        scatter_k<<<(EE * 32 + blk - 1) / blk, blk, 0, stream>>>(
            Z1 + (size_t)t * NN * HID, XW1, dinv + t * NN,
            eis + (size_t)t * 2 * EE, eis + (size_t)t * 2 * EE + EE);
    }
    relu_k<<<(TT * NN * HID + blk - 1) / blk, blk, 0, stream>>>(Z1, TT * NN * HID);

    // H2 = Z1 @ w2  (flattened: 80000 rows)
    wmma_gemm_k<<<dim3((TT * NN / 16 + 15) / 16, HID / 64), blk, 0, stream>>>(
        Z1, w2, nullptr, H2, TT * NN, HID, HID, 1, 0);

    // layer 2
    for (int t = 0; t < TT; ++t) {
        init_self_k<<<(NN * HID + blk - 1) / blk, blk, 0, stream>>>(
            Z2 + (size_t)t * NN * HID, H2 + (size_t)t * NN * HID, dinv + t * NN, b2);
        scatter_k<<<(EE * 32 + blk - 1) / blk, blk, 0, stream>>>(
            Z2 + (size_t)t * NN * HID, H2 + (size_t)t * NN * HID, dinv + t * NN,
            eis + (size_t)t * 2 * EE, eis + (size_t)t * 2 * EE + EE);
    }
    relu_k<<<(TT * NN * HID + blk - 1) / blk, blk, 0, stream>>>(Z2, TT * NN * HID);

    // GI = Z2 @ wih^T + bih   (B(k,c) = wih[c*128 + k] -> rsB=1, csB=128)
    wmma_gemm_k<<<dim3((TT * NN / 16 + 15) / 16, G3 / 64), blk, 0, stream>>>(
        Z2, wih, bih, GI, TT * NN, G3, 1, HID, 0);

    // sequential GRU scan (one WGP, big LDS)
    size_t shmem = (size_t)(64 * WPS + 16 * S_H + 4 * G3) * sizeof(float);
    hipFuncSetAttribute((const void*)gru_scan_k,
                        hipFuncAttributeMaxDynamicSharedMemorySize, (int)shmem);
    gru_scan_k<<<1, 768, shmem, stream>>>(GI, whh, bhh, out);
}